// VMD_27144193311245
// MI455X (gfx1250) — compile-verified
//
#include <hip/hip_runtime.h>

// ---------------------------------------------------------------------------
// VMD (Variational Mode Decomposition) for MI455X / gfx1250.
//   T = 2^22 samples, K = 8 modes, 30 ADMM iterations.
// Pipeline:
//   1) forward FFT of x (mixed-radix Stockham: 5x radix-16 [WMMA] + 1x radix-4)
//   2) 30x { fused streaming update kernel + deterministic finalize kernel }
//   3) per mode k: inverse FFT of u_k (spectrum computed on the fly) -> d_out
// All state lives in d_ws; everything is re-initialized each call.
// Roofline: ~24 GFLOP vs ~9 GB streamed (L2-resident; 192 MB L2, 23.3 TB/s
// HBM) => bandwidth/latency bound. Radix-16 butterflies run on the f32 WMMA
// path; streaming kernels use global_prefetch_b8 to hide L2 latency.
// ---------------------------------------------------------------------------

#define T_N      (1 << 22)
#define HALF_T   (1 << 21)
#define NR16     (T_N / 16)          // gather stride for radix-16 passes
#define K_MODES  8
#define ALPHA_C  2000.0f
#define TAU_C    1e-7f
#define TAU2_C   1e-14f
#define TOL_C    1e-6f
#define PI2_C    6.28318530717958647692f

#define ITER_BLOCKS 2048
#define ITER_EPT    8                // elements per thread (2048*256*8 = 2^22)
#define NRED        18               // 8 num + 8 den + udiff_num + udiff_den

typedef float v2f __attribute__((ext_vector_type(2)));
typedef float v8f __attribute__((ext_vector_type(8)));

static __device__ __forceinline__ v8f wmma_f32(v2f a, v2f b, v8f c) {
    // D = A(16x4) * B(4x16) + C(16x16), fp32, wave32.
    return __builtin_amdgcn_wmma_f32_16x16x4_f32(
        /*neg_a=*/false, a, /*neg_b=*/false, b,
        /*c_mod=*/(short)0, c, /*reuse_a=*/false, /*reuse_b=*/false);
}

static __device__ __forceinline__ float freq_of(int i) {
    const float invT = 1.0f / (float)T_N;
    return (float)i * invT - ((i >= HALF_T) ? 1.0f : 0.0f);
}

// ---------------------------------------------------------------------------
// Radix-16 Stockham pass. One wave handles 16 consecutive butterflies j as a
// 16x16 complex DFT matmul on the tensor cores:
//   Yr = Fr*Vr + (-Fi)*Vi ; Yi = Fr*Vi + Fi*Vr   (16 v_wmma per wave)
// mode 0: complex in[] ; mode 1: real x[] (Ns==1) ; mode 2: VMD mode-k
// spectrum u_k = (f_hat - lam/2) * w_k computed on the fly (Ns==1).
// ---------------------------------------------------------------------------
__global__ void __launch_bounds__(256)
fft_r16(const float2* __restrict__ in, float2* __restrict__ out,
        int ns, int nsLog, float sgn, int mode,
        const float* __restrict__ xr,
        const float2* __restrict__ fhat, const float2* __restrict__ lam,
        const float* __restrict__ omu, int kmode)
{
    const int lane  = threadIdx.x & 31;
    const int wave  = (blockIdx.x << 3) + (threadIdx.x >> 5);
    const int col   = lane & 15;          // butterfly within the 16-group / A row
    const int halfw = lane >> 4;          // which K-pair this lane holds
    const int j     = (wave << 4) + col;  // Stockham butterfly index

    float om_m1 = 0.f, om_0 = 0.f, om_p1 = 0.f;
    if (mode == 2) {
        om_0  = omu[kmode];
        om_m1 = (kmode > 0)           ? omu[kmode - 1] : 0.f;
        om_p1 = (kmode < K_MODES - 1) ? omu[kmode + 1] : 0.f;
    }

    const int   rem  = j & (ns - 1);
    const float twsc = sgn * PI2_C / (float)(ns << 4);

    // Gather + twiddle into B-operand layout (K rows striped per lane,
    // mirroring the documented 32-bit A 16x4 layout: VGPR0 holds K=0/K=2,
    // VGPR1 holds K=1/K=3 per half-wave).
    v2f br[4], bi[4];
#pragma unroll
    for (int kk = 0; kk < 4; ++kk) {
#pragma unroll
        for (int v = 0; v < 2; ++v) {
            const int r   = (kk << 2) + (halfw << 1) + v;  // DFT input row
            const int idx = j + r * NR16;
            float re, im;
            if (mode == 1) {
                re = xr[idx]; im = 0.f;
            } else if (mode == 2) {
                const float fq = freq_of(idx);
                float d0 = fq - om_0;
                float den = d0 * d0 + TAU2_C;
                if (kmode > 0)           { float dm = fq - om_m1; den += dm * dm; }
                if (kmode < K_MODES - 1) { float dp = fq - om_p1; den += dp * dp; }
                const float w = 1.0f / (1.0f + ALPHA_C * den);
                const float2 F = fhat[idx];
                const float2 L = lam[idx];
                re = (F.x - 0.5f * L.x) * w;
                im = (F.y - 0.5f * L.y) * w;
            } else {
                const float2 vv = in[idx]; re = vv.x; im = vv.y;
            }
            if (ns > 1) {
                const int a = (r * rem) & ((ns << 4) - 1); // exact in fp32
                float s, c;
                __sincosf(twsc * (float)a, &s, &c);
                const float tr = re * c - im * s;
                const float ti = re * s + im * c;
                re = tr; im = ti;
            }
            br[kk][v] = re; bi[kk][v] = im;
        }
    }

    // DFT-16 matrix product via WMMA (A = F16 chunks built in registers).
    v8f yr = {0.f, 0.f, 0.f, 0.f, 0.f, 0.f, 0.f, 0.f};
    v8f yi = {0.f, 0.f, 0.f, 0.f, 0.f, 0.f, 0.f, 0.f};
    const int m = col;                     // A-matrix row for this lane
#pragma unroll
    for (int kk = 0; kk < 4; ++kk) {
        v2f fr, fi, fin;
#pragma unroll
        for (int v = 0; v < 2; ++v) {
            const int r  = (kk << 2) + (halfw << 1) + v;
            const int mr = (m * r) & 15;  // exact angle reduction
            float s, c;
            __sincosf(sgn * (PI2_C / 16.0f) * (float)mr, &s, &c);
            fr[v] = c; fi[v] = s; fin[v] = -s;
        }
        yr = wmma_f32(fr,  br[kk], yr);
        yr = wmma_f32(fin, bi[kk], yr);
        yi = wmma_f32(fr,  bi[kk], yi);
        yi = wmma_f32(fi,  br[kk], yi);
    }

    // Scatter (C/D layout: lane%16 = column, VGPR v = row v + 8*halfw).
    const int grp  = j >> nsLog;
    const int base = (grp << (nsLog + 4)) + rem;
#pragma unroll
    for (int v = 0; v < 8; ++v) {
        const int d = v + (halfw << 3);
        out[base + (d << nsLog)] = make_float2(yr[v], yi[v]);
    }
}

// ---------------------------------------------------------------------------
// Final radix-4 Stockham pass (Ns = 2^20). Writes complex (forward, to f_hat)
// or real*scale (inverse, straight into d_out row for mode k).
// ---------------------------------------------------------------------------
__global__ void __launch_bounds__(256)
fft_r4_final(const float2* __restrict__ in, float2* __restrict__ outc,
             float* __restrict__ outr, float sgn, float scale)
{
    const int ns = 1 << 20;
    const int j  = blockIdx.x * 256 + threadIdx.x;   // [0, 2^20)

    float2 v0 = in[j];
    float2 v1 = in[j + ns];
    float2 v2 = in[j + 2 * ns];
    float2 v3 = in[j + 3 * ns];

    const float tws = sgn * PI2_C / (float)(ns << 2);
#pragma unroll
    for (int r = 1; r < 4; ++r) {
        float s, c;
        __sincosf(tws * (float)r * (float)j, &s, &c);
        float2* v = (r == 1) ? &v1 : (r == 2) ? &v2 : &v3;
        const float tr = v->x * c - v->y * s;
        const float ti = v->x * s + v->y * c;
        v->x = tr; v->y = ti;
    }

    const float2 t0 = make_float2(v0.x + v2.x, v0.y + v2.y);
    const float2 t1 = make_float2(v0.x - v2.x, v0.y - v2.y);
    const float2 t2 = make_float2(v1.x + v3.x, v1.y + v3.y);
    const float2 t3 = make_float2(v1.x - v3.x, v1.y - v3.y);

    const float2 y0 = make_float2(t0.x + t2.x, t0.y + t2.y);
    const float2 y2 = make_float2(t0.x - t2.x, t0.y - t2.y);
    const float2 y1 = make_float2(t1.x - sgn * t3.y, t1.y + sgn * t3.x);
    const float2 y3 = make_float2(t1.x + sgn * t3.y, t1.y - sgn * t3.x);

    if (outr) {
        outr[j]          = y0.x * scale;
        outr[j + ns]     = y1.x * scale;
        outr[j + 2 * ns] = y2.x * scale;
        outr[j + 3 * ns] = y3.x * scale;
    } else {
        outc[j]          = y0;
        outc[j + ns]     = y1;
        outc[j + 2 * ns] = y2;
        outc[j + 3 * ns] = y3;
    }
}

// ---------------------------------------------------------------------------
// Init: zero lambda, load omega_init, clear done flag.
// ---------------------------------------------------------------------------
__global__ void __launch_bounds__(256)
vmd_init(float2* __restrict__ lam, const float* __restrict__ om_init,
         float* __restrict__ omega, float* __restrict__ omega_us,
         int* __restrict__ done)
{
    const int t = blockIdx.x * 256 + threadIdx.x;
#pragma unroll
    for (int e = 0; e < ITER_EPT; ++e)
        lam[t + e * (ITER_BLOCKS * 256)] = make_float2(0.f, 0.f);
    if (blockIdx.x == 0) {
        if (threadIdx.x < K_MODES) {
            omega[threadIdx.x]    = om_init[threadIdx.x];
            omega_us[threadIdx.x] = om_init[threadIdx.x];
        }
        if (threadIdx.x == 0) *done = 0;
    }
}

// ---------------------------------------------------------------------------
// One fused ADMM iteration: streams f_hat + lam_cur (+ prev-carry snapshot),
// produces lam_next, snapshots lam_usrc, and emits fixed-order block partials
// for { omega numerators/denominators, u_diff numerator/denominator }.
// u_prev is recomputed from (omega_usrc, lam_usrc) -- never materialized.
// Successive elements per thread are 4 MiB apart; global_prefetch_b8 hides
// the L2 hit latency behind the K=8 weight arithmetic.
// ---------------------------------------------------------------------------
__global__ void __launch_bounds__(256)
vmd_iter(const float2* __restrict__ fhat,
         const float2* __restrict__ lam_cur,
         float2* __restrict__ lam_next,
         float2* __restrict__ lam_usrc,
         const float* __restrict__ omega,
         const float* __restrict__ omega_us,
         const int* __restrict__ done,
         float* __restrict__ partials,
         int first)
{
    const int tid = threadIdx.x;
    __shared__ float s_om[K_MODES], s_omu[K_MODES];
    __shared__ int   s_done;
    __shared__ float red[256];
    if (tid == 0) s_done = *done;
    if (tid < K_MODES) { s_om[tid] = omega[tid]; s_omu[tid] = omega_us[tid]; }
    __syncthreads();

    const int stride = ITER_BLOCKS * 256;
    const int i0     = blockIdx.x * 256 + tid;

    if (s_done) {  // frozen: keep ping-pong consistent, leave snapshot alone
#pragma unroll
        for (int e = 0; e < ITER_EPT; ++e) {
            const int i = i0 + e * stride;
            if (e + 1 < ITER_EPT)
                __builtin_prefetch(&lam_cur[i + stride], 0, 3);
            lam_next[i] = lam_cur[i];
        }
        return;
    }

    float om[K_MODES], omu[K_MODES];
#pragma unroll
    for (int k = 0; k < K_MODES; ++k) { om[k] = s_om[k]; omu[k] = s_omu[k]; }

    float num[K_MODES] = {0.f}, den[K_MODES] = {0.f};
    float udn = 0.f, udd = 0.f;

    // Warm the pipe for the first stripe.
    __builtin_prefetch(&fhat[i0], 0, 3);
    __builtin_prefetch(&lam_cur[i0], 0, 3);
    if (!first) __builtin_prefetch(&lam_usrc[i0], 0, 3);

    for (int e = 0; e < ITER_EPT; ++e) {
        const int i = i0 + e * stride;
        if (e + 1 < ITER_EPT) {
            const int ip = i + stride;
            __builtin_prefetch(&fhat[ip], 0, 3);
            __builtin_prefetch(&lam_cur[ip], 0, 3);
            if (!first) __builtin_prefetch(&lam_usrc[ip], 0, 3);
        }

        const float  fq = freq_of(i);
        const float2 F  = fhat[i];
        const float2 L  = lam_cur[i];
        const float  cr = F.x - 0.5f * L.x;
        const float  ci = F.y - 0.5f * L.y;
        const float  cm = cr * cr + ci * ci;

        float cpr = 0.f, cpi = 0.f, cpm = 0.f;
        if (!first) {
            const float2 Lp = lam_usrc[i];
            cpr = F.x - 0.5f * Lp.x;
            cpi = F.y - 0.5f * Lp.y;
            cpm = cpr * cpr + cpi * cpi;
        }
        lam_usrc[i] = L;   // snapshot carry-in lambda for u-recompute
        const float dotc = cr * cpr + ci * cpi;

        float sq[K_MODES], sqp[K_MODES];
#pragma unroll
        for (int k = 0; k < K_MODES; ++k) {
            const float d = fq - om[k];   sq[k]  = d * d;
            const float dp = fq - omu[k]; sqp[k] = dp * dp;
        }

        float wsum = 0.f;
#pragma unroll
        for (int k = 0; k < K_MODES; ++k) {
            float dnm = sq[k] + TAU2_C;
            if (k > 0)           dnm += sq[k - 1];
            if (k < K_MODES - 1) dnm += sq[k + 1];
            const float w = 1.0f / (1.0f + ALPHA_C * dnm);
            wsum += w;
            const float p = cm * w * w;
            num[k] += fq * p;
            den[k] += p;
            if (first) {
                udn += p;             // |u - 0|^2
            } else {
                float dnp = sqp[k] + TAU2_C;
                if (k > 0)           dnp += sqp[k - 1];
                if (k < K_MODES - 1) dnp += sqp[k + 1];
                const float wp = 1.0f / (1.0f + ALPHA_C * dnp);
                const float pp = cpm * wp * wp;
                udn += p + pp - 2.0f * w * wp * dotc;
                udd += pp;
            }
        }

        lam_next[i] = make_float2(L.x + TAU_C * (cr * wsum - F.x),
                                  L.y + TAU_C * (ci * wsum - F.y));
    }

    // Fixed-order block reduction of the 18 accumulators.
    float acc[NRED];
#pragma unroll
    for (int k = 0; k < K_MODES; ++k) { acc[k] = num[k]; acc[K_MODES + k] = den[k]; }
    acc[16] = udn; acc[17] = udd;
    for (int q = 0; q < NRED; ++q) {
        red[tid] = acc[q];
        __syncthreads();
        for (int st = 128; st > 0; st >>= 1) {
            if (tid < st) red[tid] += red[tid + st];
            __syncthreads();
        }
        if (tid == 0) partials[blockIdx.x * NRED + q] = red[0];
        __syncthreads();
    }
}

// ---------------------------------------------------------------------------
// Finalize: deterministic tree-reduce partials, update omega (+snapshot),
// evaluate convergence (n%5==0 && u_diff<TOL), set done flag.
// ---------------------------------------------------------------------------
__global__ void __launch_bounds__(256)
vmd_finalize(float* __restrict__ omega, float* __restrict__ omega_us,
             int* __restrict__ done, const float* __restrict__ partials,
             int nblk, int n)
{
    if (*done) return;
    const int tid = threadIdx.x;
    __shared__ float red[256];
    __shared__ float fin[NRED];
    for (int q = 0; q < NRED; ++q) {
        float s = 0.f;
        for (int b = tid; b < nblk; b += 256) s += partials[b * NRED + q];
        red[tid] = s;
        __syncthreads();
        for (int st = 128; st > 0; st >>= 1) {
            if (tid < st) red[tid] += red[tid + st];
            __syncthreads();
        }
        if (tid == 0) fin[q] = red[0];
        __syncthreads();
    }
    if (tid == 0) {
        const float udiff = fin[16] / (fin[17] + 1e-8f);
        const bool  conv  = ((n % 5) == 0) && (udiff < TOL_C);
        for (int k = 0; k < K_MODES; ++k) {
            omega_us[k] = omega[k];                         // carry-in snapshot
            omega[k]    = fin[k] / (fin[K_MODES + k] + 1e-8f);
        }
        if (conv) *done = 1;
    }
}

// ---------------------------------------------------------------------------
extern "C" void kernel_launch(void* const* d_in, const int* in_sizes, int n_in,
                              void* d_out, int out_size, void* d_ws, size_t ws_size,
                              hipStream_t stream)
{
    (void)in_sizes; (void)n_in; (void)out_size; (void)ws_size;
    const float* x   = (const float*)d_in[0];
    const float* om0 = (const float*)d_in[1];
    float*       out = (float*)d_out;

    char* ws = (char*)d_ws;
    const size_t SZ = (size_t)T_N * sizeof(float2);     // 32 MiB per plane
    float2* fhat = (float2*)(ws + 0 * SZ);
    float2* bufA = (float2*)(ws + 1 * SZ);              // also lambda ping
    float2* bufB = (float2*)(ws + 2 * SZ);              // also lambda pong
    float2* lamU = (float2*)(ws + 3 * SZ);              // lambda carry snapshot
    float*  omega    = (float*)(ws + 4 * SZ);
    float*  omegaU   = omega + K_MODES;
    int*    done     = (int*)(omega + 2 * K_MODES);
    float*  partials = (float*)(ws + 4 * SZ + 256);

    const dim3 b256(256);

    // ---- forward FFT of x:  5x radix-16 (WMMA) + radix-4 -> fhat ----
    fft_r16<<<2048, b256, 0, stream>>>(nullptr, bufA, 1,     0,  -1.f, 1, x, nullptr, nullptr, nullptr, 0);
    fft_r16<<<2048, b256, 0, stream>>>(bufA, bufB, 16,    4,  -1.f, 0, nullptr, nullptr, nullptr, nullptr, 0);
    fft_r16<<<2048, b256, 0, stream>>>(bufB, bufA, 256,   8,  -1.f, 0, nullptr, nullptr, nullptr, nullptr, 0);
    fft_r16<<<2048, b256, 0, stream>>>(bufA, bufB, 4096,  12, -1.f, 0, nullptr, nullptr, nullptr, nullptr, 0);
    fft_r16<<<2048, b256, 0, stream>>>(bufB, bufA, 65536, 16, -1.f, 0, nullptr, nullptr, nullptr, nullptr, 0);
    fft_r4_final<<<4096, b256, 0, stream>>>(bufA, fhat, nullptr, -1.f, 1.0f);

    // ---- ADMM state init (bufA becomes lambda=0 ping buffer) ----
    vmd_init<<<ITER_BLOCKS, b256, 0, stream>>>(bufA, om0, omega, omegaU, done);

    // ---- 30 fixed iterations with device-side freeze ----
    for (int n = 0; n < 30; ++n) {
        float2* cur = (n & 1) ? bufB : bufA;
        float2* nxt = (n & 1) ? bufA : bufB;
        vmd_iter<<<ITER_BLOCKS, b256, 0, stream>>>(fhat, cur, nxt, lamU,
                                                   omega, omegaU, done, partials,
                                                   (n == 0) ? 1 : 0);
        vmd_finalize<<<1, b256, 0, stream>>>(omega, omegaU, done, partials,
                                             ITER_BLOCKS, n);
    }

    // ---- per-mode inverse FFT; u_k recomputed on the fly in pass 0 ----
    for (int k = 0; k < K_MODES; ++k) {
        fft_r16<<<2048, b256, 0, stream>>>(nullptr, bufA, 1,     0,  1.f, 2, nullptr, fhat, lamU, omegaU, k);
        fft_r16<<<2048, b256, 0, stream>>>(bufA, bufB, 16,    4,  1.f, 0, nullptr, nullptr, nullptr, nullptr, 0);
        fft_r16<<<2048, b256, 0, stream>>>(bufB, bufA, 256,   8,  1.f, 0, nullptr, nullptr, nullptr, nullptr, 0);
        fft_r16<<<2048, b256, 0, stream>>>(bufA, bufB, 4096,  12, 1.f, 0, nullptr, nullptr, nullptr, nullptr, 0);
        fft_r16<<<2048, b256, 0, stream>>>(bufB, bufA, 65536, 16, 1.f, 0, nullptr, nullptr, nullptr, nullptr, 0);
        fft_r4_final<<<4096, b256, 0, stream>>>(bufA, nullptr, out + (size_t)k * T_N,
                                                1.f, 1.0f / (float)T_N);
    }
}